// LinkPredictionModel_73100343378050
// MI455X (gfx1250) — compile-verified
//
#include <hip/hip_runtime.h>
#include <hip/hip_bf16.h>
#include <math.h>

// ---------------------------------------------------------------------------
// CDNA5 (gfx1250) full-fp32 WMMA pipeline for the GNN link-prediction model.
// All GEMMs use V_WMMA_F32_16X16X4_F32 (exact fp32 tensor op, wave32).
// ---------------------------------------------------------------------------

typedef __attribute__((ext_vector_type(2))) float v2f;
typedef __attribute__((ext_vector_type(8))) float v8f;

// Guard only in the device pass: host clang reports target builtins as absent.
#if defined(__HIP_DEVICE_COMPILE__) && \
    !__has_builtin(__builtin_amdgcn_wmma_f32_16x16x4_f32)
#error "wmma f32 16x16x4 builtin missing on this toolchain"
#endif

static __device__ __forceinline__ v8f wmma4(v2f a, v2f b, v8f c) {
  // 8 args: (neg_a, A, neg_b, B, c_mod, C, reuse_a, reuse_b)
  return __builtin_amdgcn_wmma_f32_16x16x4_f32(false, a, false, b, (short)0, c,
                                               false, false);
}

static __device__ __forceinline__ float sigmoidf(float v) {
  return 1.0f / (1.0f + __expf(-v));
}

#define H 64

// ---------------------------------------------------------------------------
// Kernel 1: h = x @ Wproj + bproj   [N,1024]x[1024,64]
// One wave per 16x16 output tile; A chunks (16x32) staged via LDS float4s.
// ---------------------------------------------------------------------------
__global__ void k_proj(const float* __restrict__ x,
                       const float* __restrict__ Wproj,
                       const float* __restrict__ bproj,
                       float* __restrict__ h, int nRowTiles) {
  __shared__ __align__(16) float lds[8][16 * 36];  // stride 36: 16B-aligned, conflict-free
  const int wave = threadIdx.x >> 5;
  const int lane = threadIdx.x & 31;
  const int task = blockIdx.x * 8 + wave;
  if (task >= nRowTiles * 4) return;                // wave-uniform guard
  const int rowTile = task >> 2, colTile = task & 3;
  const int m = lane & 15, half = lane >> 4, kb = half * 2;
  const int colBase = colTile * 16;
  float* at = lds[wave];
  const float* xrow = x + (size_t)(rowTile * 16 + m) * 1024;
  v8f acc = {};
  for (int k0 = 0; k0 < 1024; k0 += 32) {
    // stage 16x32 fp32 chunk of x (row m, 16 cols per half-lane-group)
    const float4* s4 = (const float4*)(xrow + k0 + half * 16);
    float4* d4 = (float4*)(at + m * 36 + half * 16);
    d4[0] = s4[0]; d4[1] = s4[1]; d4[2] = s4[2]; d4[3] = s4[3];
#pragma unroll
    for (int kk = 0; kk < 32; kk += 4) {
      v2f a, b;
      a.x = at[m * 36 + kk + kb];
      a.y = at[m * 36 + kk + kb + 1];
      const float* bp = Wproj + (size_t)(k0 + kk + kb) * H + colBase + m;
      b.x = bp[0];
      b.y = bp[H];
      acc = wmma4(a, b, acc);
    }
  }
  const float bias = bproj[colBase + m];
  float* out = h + (size_t)(rowTile * 16) * H + colBase + m;
#pragma unroll
  for (int r = 0; r < 8; ++r)
    out[(size_t)(r + 8 * half) * H] = acc[r] + bias;
}

// ---------------------------------------------------------------------------
// Kernel 2: both layers' edge gates in ONE stream over q_expand/edge_attr.
// gate[l][e] = sigmoid( dot(q_expand[e]*edge_attr[e], wphi[l]) + bphi[l] )
// One wave per edge, coalesced 2x32-lane loads, shuffle reduction.
// ---------------------------------------------------------------------------
__global__ void k_gate(const float* __restrict__ qe, const float* __restrict__ ea,
                       const float* __restrict__ wphi, const float* __restrict__ bphi,
                       float* __restrict__ gate, int E) {
  const int wave = threadIdx.x >> 5, lane = threadIdx.x & 31;
  const int e = blockIdx.x * 8 + wave;
  if (e >= E) return;
  const size_t base = (size_t)e * H;
  const float p0 = qe[base + lane] * ea[base + lane];
  const float p1 = qe[base + lane + 32] * ea[base + lane + 32];
  float a0 = p0 * wphi[lane] + p1 * wphi[lane + 32];
  float a1 = p0 * wphi[64 + lane] + p1 * wphi[64 + lane + 32];
#pragma unroll
  for (int off = 16; off > 0; off >>= 1) {
    a0 += __shfl_xor(a0, off, 32);
    a1 += __shfl_xor(a1, off, 32);
  }
  if (lane == 0) {
    gate[e] = sigmoidf(a0 + bphi[0]);
    gate[(size_t)E + e] = sigmoidf(a1 + bphi[1]);
  }
}

// ---------------------------------------------------------------------------
// Kernel 3: msg = gate * (h[src] @ Wm + bm); atomic scatter-add into aggr[dst].
// One wave per 16-edge tile; gathered h rows staged in LDS (L2-resident h).
// ---------------------------------------------------------------------------
__global__ void k_msg(const float* __restrict__ h, const int* __restrict__ eidx,
                      const float* __restrict__ Wm, const float* __restrict__ bm,
                      const float* __restrict__ gate, float* __restrict__ aggr,
                      int E) {
  __shared__ __align__(16) float ldsA[8][16 * 68];
  __shared__ float ldsG[8][16];
  __shared__ int ldsD[8][16];
  const int wave = threadIdx.x >> 5, lane = threadIdx.x & 31;
  const int tile = blockIdx.x * 8 + wave;
  if (tile >= (E >> 4)) return;
  const int base = tile * 16;
  const int m = lane & 15, half = lane >> 4, kb = half * 2;
  {
    const int srcIdx = eidx[base + m];             // edge_index[0]
    if (half == 0) {
      ldsG[wave][m] = gate[base + m];
      ldsD[wave][m] = eidx[E + base + m];          // edge_index[1]
    }
    const float4* s4 = (const float4*)(h + (size_t)srcIdx * H + half * 32);
    float4* d4 = (float4*)(&ldsA[wave][m * 68 + half * 32]);
    d4[0] = s4[0]; d4[1] = s4[1]; d4[2] = s4[2]; d4[3] = s4[3];
  }
  const float* at = ldsA[wave];
#pragma unroll
  for (int ct = 0; ct < 4; ++ct) {
    v8f acc = {};
    const int colBase = ct * 16;
#pragma unroll
    for (int k = 0; k < H; k += 4) {
      v2f a, b;
      a.x = at[m * 68 + k + kb];
      a.y = at[m * 68 + k + kb + 1];
      const float* bp = Wm + (size_t)(k + kb) * H + colBase + m;
      b.x = bp[0];
      b.y = bp[H];
      acc = wmma4(a, b, acc);
    }
    const float bias = bm[colBase + m];
#pragma unroll
    for (int r = 0; r < 8; ++r) {
      const int M = r + 8 * half;
      const float val = (acc[r] + bias) * ldsG[wave][M];
      atomicAdd(&aggr[(size_t)ldsD[wave][M] * H + colBase + m], val);
    }
  }
}

// ---------------------------------------------------------------------------
// Kernel 4: GRU cell, in-place h update (row-local, race-free).
// gi = aggr@Wih^T, gh = h@Whh^T; 6 fp32 WMMA accumulators per 16-col tile.
// ---------------------------------------------------------------------------
__global__ void k_gru(float* __restrict__ h, const float* __restrict__ aggr,
                      const float* __restrict__ Wih, const float* __restrict__ Whh,
                      const float* __restrict__ bih, const float* __restrict__ bhh,
                      int N) {
  __shared__ __align__(16) float ldsA[8][16 * 68];
  __shared__ __align__(16) float ldsH[8][16 * 68];
  const int wave = threadIdx.x >> 5, lane = threadIdx.x & 31;
  const int tile = blockIdx.x * 8 + wave;
  if (tile >= (N >> 4)) return;
  const int base = tile * 16;
  const int m = lane & 15, half = lane >> 4, kb = half * 2;
  {
    const float4* sa = (const float4*)(aggr + (size_t)(base + m) * H + half * 32);
    const float4* sh = (const float4*)(h + (size_t)(base + m) * H + half * 32);
    float4* da = (float4*)(&ldsA[wave][m * 68 + half * 32]);
    float4* dh = (float4*)(&ldsH[wave][m * 68 + half * 32]);
    da[0] = sa[0]; da[1] = sa[1]; da[2] = sa[2]; da[3] = sa[3];
    dh[0] = sh[0]; dh[1] = sh[1]; dh[2] = sh[2]; dh[3] = sh[3];
  }
  const float* at = ldsA[wave];
  const float* ht = ldsH[wave];
#pragma unroll
  for (int ct = 0; ct < 4; ++ct) {
    v8f ai_r = {}, ah_r = {}, ai_z = {}, ah_z = {}, ai_n = {}, ah_n = {};
    const int col = ct * 16 + m;  // GRU output column == weight row j
    for (int k = 0; k < H; k += 4) {
      v2f a1, a2, b;
      a1.x = at[m * 68 + k + kb]; a1.y = at[m * 68 + k + kb + 1];
      a2.x = ht[m * 68 + k + kb]; a2.y = ht[m * 68 + k + kb + 1];
      // B[k][j] = W[j][k]  (W stored [3H,H] row-major)
      b.x = Wih[(size_t)col * H + k + kb];        b.y = Wih[(size_t)col * H + k + kb + 1];
      ai_r = wmma4(a1, b, ai_r);
      b.x = Whh[(size_t)col * H + k + kb];        b.y = Whh[(size_t)col * H + k + kb + 1];
      ah_r = wmma4(a2, b, ah_r);
      b.x = Wih[(size_t)(64 + col) * H + k + kb]; b.y = Wih[(size_t)(64 + col) * H + k + kb + 1];
      ai_z = wmma4(a1, b, ai_z);
      b.x = Whh[(size_t)(64 + col) * H + k + kb]; b.y = Whh[(size_t)(64 + col) * H + k + kb + 1];
      ah_z = wmma4(a2, b, ah_z);
      b.x = Wih[(size_t)(128 + col) * H + k + kb]; b.y = Wih[(size_t)(128 + col) * H + k + kb + 1];
      ai_n = wmma4(a1, b, ai_n);
      b.x = Whh[(size_t)(128 + col) * H + k + kb]; b.y = Whh[(size_t)(128 + col) * H + k + kb + 1];
      ah_n = wmma4(a2, b, ah_n);
    }
    const float br_i = bih[col], br_h = bhh[col];
    const float bz_i = bih[64 + col], bz_h = bhh[64 + col];
    const float bn_i = bih[128 + col], bn_h = bhh[128 + col];
#pragma unroll
    for (int r = 0; r < 8; ++r) {
      const int M = r + 8 * half;
      const float rr = sigmoidf(ai_r[r] + br_i + ah_r[r] + br_h);
      const float zz = sigmoidf(ai_z[r] + bz_i + ah_z[r] + bz_h);
      const float nn = tanhf(ai_n[r] + bn_i + rr * (ah_n[r] + bn_h));
      const float ho = ht[M * 68 + ct * 16 + m];
      h[(size_t)(base + M) * H + ct * 16 + m] = (1.0f - zz) * nn + zz * ho;
    }
  }
}

// ---------------------------------------------------------------------------
// Kernel 5: pair scoring MLP + BCEWithLogits mean (atomic accumulate).
// feat [16x256] built in LDS, hidden via WMMA vs W1, W2 dot via shuffles.
// ---------------------------------------------------------------------------
__global__ void k_pair(const float* __restrict__ h, const int* __restrict__ src,
                       const int* __restrict__ dst, const float* __restrict__ labels,
                       const float* __restrict__ qp, const float* __restrict__ W1,
                       const float* __restrict__ b1, const float* __restrict__ W2,
                       const float* __restrict__ b2, float* __restrict__ out, int P) {
  __shared__ __align__(16) float lds[4][16 * 260];
  const int wave = threadIdx.x >> 5, lane = threadIdx.x & 31;
  const int tile = blockIdx.x * 4 + wave;
  if (tile >= (P >> 4)) return;
  const int base = tile * 16;
  const int m = lane & 15, half = lane >> 4, kb = half * 2;
  float* ft = lds[wave];
  for (int idx = lane; idx < 16 * 256; idx += 32) {
    const int row = idx >> 8, c = idx & 255;
    const int s = src[base + row], d = dst[base + row];
    float v;
    if (c < 64)       v = h[(size_t)s * H + c];
    else if (c < 128) v = h[(size_t)d * H + (c - 64)];
    else if (c < 192) v = h[(size_t)s * H + (c - 128)] * h[(size_t)d * H + (c - 128)];
    else              v = qp[(size_t)(base + row) * H + (c - 192)];
    ft[row * 260 + c] = v;
  }
  v8f hid[4];
#pragma unroll
  for (int ct = 0; ct < 4; ++ct) {
    v8f acc = {};
    for (int k = 0; k < 256; k += 4) {
      v2f a, b;
      a.x = ft[m * 260 + k + kb];
      a.y = ft[m * 260 + k + kb + 1];
      const float* bp = W1 + (size_t)(k + kb) * H + ct * 16 + m;
      b.x = bp[0];
      b.y = bp[H];
      acc = wmma4(a, b, acc);
    }
    hid[ct] = acc;
  }
  const float inv = 1.0f / (float)P;
#pragma unroll
  for (int r = 0; r < 8; ++r) {
    float partial = 0.0f;
#pragma unroll
    for (int ct = 0; ct < 4; ++ct) {
      float v = fmaxf(hid[ct][r] + b1[ct * 16 + m], 0.0f);
      partial += v * W2[ct * 16 + m];
    }
    partial += __shfl_xor(partial, 1, 32);
    partial += __shfl_xor(partial, 2, 32);
    partial += __shfl_xor(partial, 4, 32);
    partial += __shfl_xor(partial, 8, 32);
    if (m == 0) {  // lanes 0 and 16 each own 8 rows
      const int M = r + 8 * half;
      const float logit = partial + b2[0];
      const float y = labels[base + M];
      const float term =
          fmaxf(logit, 0.0f) - logit * y + log1pf(__expf(-fabsf(logit)));
      atomicAdd(out, term * inv);
    }
  }
}

// ---------------------------------------------------------------------------
extern "C" void kernel_launch(void* const* d_in, const int* in_sizes, int n_in,
                              void* d_out, int out_size, void* d_ws, size_t ws_size,
                              hipStream_t stream) {
  const float* x      = (const float*)d_in[0];
  const int*   eidx   = (const int*)d_in[1];   // edge_index [2,E]
  const float* ea     = (const float*)d_in[2];
  const float* qe     = (const float*)d_in[3];
  const int*   src    = (const int*)d_in[4];
  const int*   dst    = (const int*)d_in[5];
  const float* labels = (const float*)d_in[6];
  const float* qp     = (const float*)d_in[7];
  const float* Wproj  = (const float*)d_in[8];
  const float* bproj  = (const float*)d_in[9];
  const float* Wm     = (const float*)d_in[10];
  const float* bm     = (const float*)d_in[11];
  const float* wphi   = (const float*)d_in[12];
  const float* bphi   = (const float*)d_in[13];
  const float* Wih    = (const float*)d_in[14];
  const float* Whh    = (const float*)d_in[15];
  const float* bih    = (const float*)d_in[16];
  const float* bhh    = (const float*)d_in[17];
  const float* W1     = (const float*)d_in[18];
  const float* b1     = (const float*)d_in[19];
  const float* W2     = (const float*)d_in[20];
  const float* b2     = (const float*)d_in[21];

  const int N = in_sizes[0] / 1024;
  const int E = in_sizes[2] / H;
  const int P = in_sizes[4];

  // workspace: h [N*64] | aggr [N*64] | gate [2*E]   (~64 MB total)
  float* hbuf = (float*)d_ws;
  float* aggr = hbuf + (size_t)N * H;
  float* gate = hbuf + (size_t)N * H * 2;

  // both layers' edge gates in one stream over the 820 MB of edge features
  k_gate<<<(E + 7) / 8, 256, 0, stream>>>(qe, ea, wphi, bphi, gate, E);
  // input projection (HBM-bound 410 MB stream, fp32 WMMA)
  k_proj<<<((N / 16) * 4 + 7) / 8, 256, 0, stream>>>(x, Wproj, bproj, hbuf, N / 16);

  for (int l = 0; l < 2; ++l) {
    (void)hipMemsetAsync(aggr, 0, (size_t)N * H * sizeof(float), stream);
    k_msg<<<((E / 16) + 7) / 8, 256, 0, stream>>>(
        hbuf, eidx, Wm + l * H * H, bm + l * H, gate + (size_t)l * E, aggr, E);
    k_gru<<<((N / 16) + 7) / 8, 256, 0, stream>>>(
        hbuf, aggr, Wih + l * 3 * H * H, Whh + l * 3 * H * H,
        bih + l * 3 * H, bhh + l * 3 * H, N);
  }

  (void)hipMemsetAsync(d_out, 0, sizeof(float), stream);
  k_pair<<<((P / 16) + 3) / 4, 128, 0, stream>>>(hbuf, src, dst, labels, qp, W1,
                                                 b1, W2, b2, (float*)d_out, P);
}